// Histogram_3547642986677
// MI455X (gfx1250) — compile-verified
//
#include <hip/hip_runtime.h>

#define NBINS   64
#define NCOPIES 16          // one private histogram per half-wave
#define ROWLEN  8192
#define THREADS 256         // 8 wave32 per block
#define VMIN_F  (-3.0f)
#define VMAX_F  (3.0f)

typedef __attribute__((ext_vector_type(2))) float v2f;
typedef __attribute__((ext_vector_type(8))) float v8f;

__device__ __forceinline__ int bin_of(float v) {
    // Same semantics as tf.histogram_fixed_width: scale into [0, NBINS), clip to edges.
    const float scale = (float)NBINS / (VMAX_F - VMIN_F);
    int b = (int)floorf((v - VMIN_F) * scale);
    b = b < 0 ? 0 : b;
    b = b > (NBINS - 1) ? (NBINS - 1) : b;
    return b;
}

__global__ __launch_bounds__(THREADS) void hist_rows_kernel(
    const float* __restrict__ x, float* __restrict__ out, int rowlen, float inv_n) {
    __shared__ float hist[NCOPIES][NBINS];   // 4 KB, per-half-wave private copies

    const int tid = threadIdx.x;
    const int row = blockIdx.x;

    // Zero the sub-histograms.
    for (int i = tid; i < NCOPIES * NBINS; i += THREADS)
        ((float*)hist)[i] = 0.0f;
    __syncthreads();

    // Phase 1: streaming binning. float4 loads -> global_load_b128, coalesced.
    const int copy = tid >> 4;               // 16 lanes share one private copy
    const float4* in4 = (const float4*)(x + (size_t)row * rowlen);
    const int iters = rowlen / 4 / THREADS;  // 8
    #pragma unroll
    for (int i = 0; i < iters; ++i) {
        float4 v = in4[(size_t)i * THREADS + tid];
        atomicAdd(&hist[copy][bin_of(v.x)], 1.0f);
        atomicAdd(&hist[copy][bin_of(v.y)], 1.0f);
        atomicAdd(&hist[copy][bin_of(v.z)], 1.0f);
        atomicAdd(&hist[copy][bin_of(v.w)], 1.0f);
    }
    __syncthreads();

    // Phase 2: reduce 16 sub-histograms with V_WMMA_F32_16X16X4_F32.
    // D[m][n] = sum_k A[m][k]*B[k][n], A = ones(16x4), B = 4 sub-hist rows x 16 bins.
    // 4 K-chunks (16 copies / 4) x 4 bin-groups (64 bins / 16) = 16 WMMAs, wave 0 only
    // (EXEC all-ones within the wave, as WMMA requires).
    if (tid < 32) {
        const int lane = tid;          // wave 0
        const int n    = lane & 15;    // bin within group
        const int hi   = lane >> 4;    // lane-half selects K=2,3 rows
        v2f a;
        a.x = 1.0f; a.y = 1.0f;        // A = ones, identical in all lanes

        #pragma unroll
        for (int g = 0; g < 4; ++g) {          // bin group: bins [16g, 16g+16)
            v8f acc = {};
            #pragma unroll
            for (int c = 0; c < 4; ++c) {      // K-chunk: copies [4c, 4c+4)
                // B 4x16 f32 layout (mirror of documented 16x4 A layout):
                //   VGPR0: lanes 0-15 = row K=0, lanes 16-31 = row K=2
                //   VGPR1: lanes 0-15 = row K=1, lanes 16-31 = row K=3
                v2f b;
                b.x = hist[4 * c + (hi ? 2 : 0)][16 * g + n];
                b.y = hist[4 * c + (hi ? 3 : 1)][16 * g + n];
                acc = __builtin_amdgcn_wmma_f32_16x16x4_f32(
                    /*neg_a=*/false, a, /*neg_b=*/false, b,
                    /*c_mod=*/(short)0, acc, /*reuse_a=*/false, /*reuse_b=*/false);
            }
            // D row M=0 lives in acc[0] on lanes 0-15 (N = lane).
            if (hi == 0)
                out[(size_t)row * NBINS + 16 * g + n] = acc[0] * inv_n;
        }
    }
}

extern "C" void kernel_launch(void* const* d_in, const int* in_sizes, int n_in,
                              void* d_out, int out_size, void* d_ws, size_t ws_size,
                              hipStream_t stream) {
    (void)n_in; (void)out_size; (void)d_ws; (void)ws_size;
    const float* x  = (const float*)d_in[0];
    float* out      = (float*)d_out;
    const int rowlen = ROWLEN;                   // N = 8192 (reference shape)
    const int rows   = in_sizes[0] / rowlen;     // B = 4096
    const float inv_n = 1.0f / (float)rowlen;    // host-side constant: no device divide
    hist_rows_kernel<<<rows, THREADS, 0, stream>>>(x, out, rowlen, inv_n);
}